// Extractor_89910845375089
// MI455X (gfx1250) — compile-verified
//
#include <hip/hip_runtime.h>
#include <hip/hip_bf16.h>
#include <math.h>

typedef __attribute__((ext_vector_type(2))) float v2f;
typedef __attribute__((ext_vector_type(8))) float v8f;

#define HIDDIM 128

// ---------------------------------------------------------------- utilities
__global__ void fill_zero_kernel(float* __restrict__ p, long long n) {
  long long i = (long long)blockIdx.x * blockDim.x + threadIdx.x;
  if (i < n) p[i] = 0.0f;
}

__global__ void count_edges_kernel(const int* __restrict__ dst, int e,
                                   float* __restrict__ cnt) {
  int i = blockIdx.x * blockDim.x + threadIdx.x;
  if (i < e) atomicAdd(&cnt[dst[i]], 1.0f);
}

__global__ void invcnt_kernel(const float* __restrict__ cnt,
                              float* __restrict__ inv, int n) {
  int i = blockIdx.x * blockDim.x + threadIdx.x;
  if (i < n) inv[i] = 1.0f / fmaxf(cnt[i], 1.0f);
}

// layer-1 scatter (D=8): one thread per (edge, feature)
__global__ void scatter_add8_kernel(const float* __restrict__ x,
                                    const int* __restrict__ src,
                                    const int* __restrict__ dst,
                                    float* __restrict__ agg, long long total) {
  long long i = (long long)blockIdx.x * blockDim.x + threadIdx.x;
  if (i >= total) return;
  int e = (int)(i >> 3);
  int k = (int)(i & 7);
  atomicAdd(&agg[(size_t)dst[e] * 8 + k], x[(size_t)src[e] * 8 + k]);
}

// 128-wide scatter: one wave32 per edge; lane moves 4 contiguous floats.
// Edge id forced wave-uniform via readfirstlane so index loads scalarize.
__global__ __launch_bounds__(256) void scatter_add128_kernel(
    const float* __restrict__ x, const int* __restrict__ src,
    const int* __restrict__ dst, float* __restrict__ agg, int E) {
  int wave = (int)((blockIdx.x * blockDim.x + threadIdx.x) >> 5);
  wave = __builtin_amdgcn_readfirstlane(wave);
  if (wave >= E) return;
  const int lane = threadIdx.x & 31;
  const int s = src[wave];
  const int d = dst[wave];
  const float4 v = *(const float4*)(x + (size_t)s * HIDDIM + lane * 4);
  float* p = agg + (size_t)d * HIDDIM + lane * 4;
  atomicAdd(p + 0, v.x);
  atomicAdd(p + 1, v.y);
  atomicAdd(p + 2, v.z);
  atomicAdd(p + 3, v.w);
}

// lane-xor reduce step via DS_SWIZZLE (group-of-32 bit mode, and=0x1f):
// xor masks < 16 never cross the 16-lane halves, matching the C-layout split.
__device__ __forceinline__ float xor_add(float s, const int imm) {
  switch (imm) {  // ds_swizzle offset must be a literal constant
    case 1:  s += __int_as_float(__builtin_amdgcn_ds_swizzle(__float_as_int(s), 0x041f)); break;
    case 2:  s += __int_as_float(__builtin_amdgcn_ds_swizzle(__float_as_int(s), 0x081f)); break;
    case 4:  s += __int_as_float(__builtin_amdgcn_ds_swizzle(__float_as_int(s), 0x101f)); break;
    default: s += __int_as_float(__builtin_amdgcn_ds_swizzle(__float_as_int(s), 0x201f)); break;
  }
  return s;
}

// ------------------------------------------------ fused SAGE layer (WMMA f32)
// out[r,:] = tanh( l2norm( mean[r,:] @ wl + x[r,:] @ wr + b ) )
// block = 256 threads (8 wave32); block owns 16 rows, wave owns 16 columns.
template <int K>
__global__ __launch_bounds__(256) void sage_layer_kernel(
    const float* __restrict__ agg, const float* __restrict__ invcnt,
    const float* __restrict__ xin, const float* __restrict__ wl,
    const float* __restrict__ wr, const float* __restrict__ bias,
    float* __restrict__ out) {
  __shared__ float sM[16][K + 1];  // mean tile (padded: no LDS bank conflicts)
  __shared__ float sX[16][K + 1];  // x tile
  __shared__ float s_norm[16];

  const int tid = threadIdx.x;
  const int wave = tid >> 5;
  const int lane = tid & 31;
  const int half = lane >> 4;  // K-split half per A/B 16x4 layout
  const int l16 = lane & 15;
  const int rowBase = blockIdx.x * 16;
  const int col = wave * 16 + l16;  // output column owned by this lane

  if (tid < 16) s_norm[tid] = 0.0f;
  // cooperative, coalesced stage of both A operands (8x reuse across waves)
  for (int i = tid; i < 16 * K; i += 256) {
    int r = i / K;
    int k = i - r * K;
    size_t g = (size_t)(rowBase + r) * K + k;
    sM[r][k] = agg[g] * invcnt[rowBase + r];
    sX[r][k] = xin[g];
  }
  __syncthreads();

  // A 16x4 f32: lanes 0-15 -> M=l16, K={k0,k0+1}; lanes 16-31 -> K={k0+2,k0+3}
  // B 4x16 mirrors A with lane = N.
  v8f c = {};
#pragma unroll 8
  for (int k0 = 0; k0 < K; k0 += 4) {
    const int ka = k0 + 2 * half;
    v2f am, ax, bl, br;
    am.x = sM[l16][ka];
    am.y = sM[l16][ka + 1];
    ax.x = sX[l16][ka];
    ax.y = sX[l16][ka + 1];
    bl.x = wl[(size_t)ka * HIDDIM + col];
    bl.y = wl[(size_t)(ka + 1) * HIDDIM + col];
    br.x = wr[(size_t)ka * HIDDIM + col];
    br.y = wr[(size_t)(ka + 1) * HIDDIM + col];
    // D = A*B + C (fp32 WMMA, both GEMMs accumulate into the same C)
    c = __builtin_amdgcn_wmma_f32_16x16x4_f32(false, am, false, bl, (short)0, c,
                                              false, false);
    c = __builtin_amdgcn_wmma_f32_16x16x4_f32(false, ax, false, br, (short)0, c,
                                              false, false);
  }

  const float bv = bias[col];
#pragma unroll
  for (int v = 0; v < 8; ++v) c[v] += bv;

  // C layout: lane<16 -> (M=v, N=l16); lane>=16 -> (M=v+8, N=l16).
  // Row sum-of-squares: ds_swizzle xor-reduce over the 16 lanes of each half,
  // then combine the 8 waves via one LDS float atomic per row.
#pragma unroll
  for (int v = 0; v < 8; ++v) {
    float s = c[v] * c[v];
    s = xor_add(s, 1);
    s = xor_add(s, 2);
    s = xor_add(s, 4);
    s = xor_add(s, 8);
    if (l16 == 0) atomicAdd(&s_norm[v + 8 * half], s);
  }
  __syncthreads();

#pragma unroll
  for (int v = 0; v < 8; ++v) {
    const int row = v + 8 * half;
    const float scale = 1.0f / fmaxf(sqrtf(s_norm[row]), 1e-12f);
    out[(size_t)(rowBase + row) * HIDDIM + col] = tanhf(c[v] * scale);
  }
}

// ---------------------------------------------------------------- launcher
extern "C" void kernel_launch(void* const* d_in, const int* in_sizes, int n_in,
                              void* d_out, int out_size, void* d_ws,
                              size_t ws_size, hipStream_t stream) {
  const float* x = (const float*)d_in[0];
  const int* ei_conn = (const int*)d_in[1];
  const int* ei_dest = (const int*)d_in[2];
  const float* wl1 = (const float*)d_in[3];
  const float* wr1 = (const float*)d_in[4];
  const float* b1 = (const float*)d_in[5];
  const float* wl2 = (const float*)d_in[6];
  const float* wr2 = (const float*)d_in[7];
  const float* b2 = (const float*)d_in[8];
  const float* wl3 = (const float*)d_in[9];
  const float* wr3 = (const float*)d_in[10];
  const float* b3 = (const float*)d_in[11];

  const int N = in_sizes[0] / 8;  // IN_DIM = 8
  const int E = in_sizes[1] / 2;  // edge_index is [2, E]

  // workspace layout (floats): agg | h1 | h2 | invc_conn | invc_dest | cnt
  float* ws = (float*)d_ws;
  float* agg = ws;
  float* h1 = agg + (size_t)N * HIDDIM;
  float* h2 = h1 + (size_t)N * HIDDIM;
  float* ivc_conn = h2 + (size_t)N * HIDDIM;
  float* ivc_dest = ivc_conn + N;
  float* cnt = ivc_dest + N;

  const int T = 256;
  auto nb = [](long long n, int t) { return (int)((n + t - 1) / t); };

  const int* src_conn = ei_conn;
  const int* dst_conn = ei_conn + E;
  const int* src_dest = ei_dest;
  const int* dst_dest = ei_dest + E;

  // degrees -> inverse counts for both edge lists
  fill_zero_kernel<<<nb(N, T), T, 0, stream>>>(cnt, N);
  count_edges_kernel<<<nb(E, T), T, 0, stream>>>(dst_conn, E, cnt);
  invcnt_kernel<<<nb(N, T), T, 0, stream>>>(cnt, ivc_conn, N);
  fill_zero_kernel<<<nb(N, T), T, 0, stream>>>(cnt, N);
  count_edges_kernel<<<nb(E, T), T, 0, stream>>>(dst_dest, E, cnt);
  invcnt_kernel<<<nb(N, T), T, 0, stream>>>(cnt, ivc_dest, N);

  const int gemmBlocks = (N + 15) / 16;       // 100000 -> 6250 (exact)
  const int scatBlocks = (E + 7) / 8;         // 8 waves (edges) per block

  // layer 1 (K = 8, conn edges)
  fill_zero_kernel<<<nb((long long)N * 8, T), T, 0, stream>>>(agg,
                                                              (long long)N * 8);
  scatter_add8_kernel<<<nb((long long)E * 8, T), T, 0, stream>>>(
      x, src_conn, dst_conn, agg, (long long)E * 8);
  sage_layer_kernel<8><<<gemmBlocks, T, 0, stream>>>(agg, ivc_conn, x, wl1,
                                                     wr1, b1, h1);

  // layer 2 (K = 128, dest edges)
  fill_zero_kernel<<<nb((long long)N * HIDDIM, T), T, 0, stream>>>(
      agg, (long long)N * HIDDIM);
  scatter_add128_kernel<<<scatBlocks, T, 0, stream>>>(h1, src_dest, dst_dest,
                                                      agg, E);
  sage_layer_kernel<128><<<gemmBlocks, T, 0, stream>>>(agg, ivc_dest, h1, wl2,
                                                       wr2, b2, h2);

  // layer 3 (K = 128, conn edges) -> d_out
  fill_zero_kernel<<<nb((long long)N * HIDDIM, T), T, 0, stream>>>(
      agg, (long long)N * HIDDIM);
  scatter_add128_kernel<<<scatBlocks, T, 0, stream>>>(h2, src_conn, dst_conn,
                                                      agg, E);
  sage_layer_kernel<128><<<gemmBlocks, T, 0, stream>>>(agg, ivc_conn, h2, wl3,
                                                       wr3, b3, (float*)d_out);
}